// Crop_12618613916200
// MI455X (gfx1250) — compile-verified
//
#include <hip/hip_runtime.h>
#include <stdint.h>

typedef __attribute__((ext_vector_type(2))) float    v2f;
typedef __attribute__((ext_vector_type(8))) float    v8f;
typedef __attribute__((ext_vector_type(4))) uint32_t u32x4;
typedef __attribute__((ext_vector_type(8))) uint32_t u32x8;

#define CROP  14
#define NCH   256
#define BATCH 4            // channels per TDM descriptor (tile_dim2)
#define NB    (32 / BATCH) // batches per wave (each wave owns 32 channels)

// Issue one Tensor-Data-Mover load of a [BATCH][16][16] f32 tile:
//   global layout: x contiguous, row stride = row_stride_elts, channel stride =
//   chan_stride_elts; LDS layout: packed [c][y][x].
// OOB (x >= rem_x, y >= rem_y, c >= rem_c) reads return zero, which carries
// zero interpolation weight by construction of Wy/Wx.
__device__ __forceinline__ void tdm_load_patch3d(
    uint32_t lds_byte_addr, const float* gaddr,
    uint32_t rem_x, uint32_t rem_y, uint32_t rem_c,
    uint32_t row_stride_elts, uint32_t chan_stride_elts)
{
    uint64_t ga = (uint64_t)(uintptr_t)gaddr;

    u32x4 g0;
    g0[0] = 1u;                                                 // count=1, user mode, no gather
    g0[1] = lds_byte_addr;                                      // lds_addr (bytes)
    g0[2] = (uint32_t)ga;                                       // global_addr[31:0]
    g0[3] = ((uint32_t)(ga >> 32) & 0x01FFFFFFu) | (2u << 30);  // global_addr[56:32] | type=2

    u32x8 g1;
    g1[0] = 2u << 16;                                   // wg_mask=0 | data_size=2 (4B) | no flags
    g1[1] = (rem_x & 0xFFFFu) << 16;                    // tensor_dim0[15:0]
    g1[2] = (rem_x >> 16) | ((rem_y & 0xFFFFu) << 16);  // tensor_dim0[31:16] | tensor_dim1[15:0]
    g1[3] = (rem_y >> 16) | (16u << 16);                // tensor_dim1[31:16] | tile_dim0=16
    g1[4] = 16u | ((uint32_t)BATCH << 16);              // tile_dim1=16 | tile_dim2=BATCH
    g1[5] = row_stride_elts;                            // tensor_dim0_stride[31:0]
    g1[6] = (chan_stride_elts & 0xFFFFu) << 16;         // stride0[47:32]=0 | stride1[15:0]
    g1[7] = chan_stride_elts >> 16;                     // stride1[47:16]

    u32x4 g2;
    g2[0] = rem_c;                                      // tensor_dim2
    g2[1] = 0u;                                         // tensor_dim3 (unused)
    g2[2] = 0u;                                         // tensor_dim2_stride lo (unused, 3D)
    g2[3] = 0u;                                         // stride hi | tile_dim3=0
    u32x4 g3 = {0u, 0u, 0u, 0u};

    asm volatile("tensor_load_to_lds %0, %1, %2, %3"
                 :: "s"(g0), "s"(g1), "s"(g2), "s"(g3)
                 : "memory");
}

// One block per ROI. 256 threads = 8 wave32s, each wave handles 32 channels.
// Per channel: Out(16x16) = Wy(16x16) * Patch(16x16) * WxT(16x16) via
// chained V_WMMA_F32_16X16X4_F32; patches arrive via double-buffered TDM DMA.
__global__ __launch_bounds__(256) void roi_crop_wmma_kernel(
    const float* __restrict__ f0, const float* __restrict__ f1,
    const float* __restrict__ f2, const float* __restrict__ f3,
    const float* __restrict__ props, float* __restrict__ out)
{
    __shared__ float Wy_s[16][16];                 // [sy][dy] (A of GEMM1), rows 14,15 zero
    __shared__ float Wx_s[16][16];                 // [dx][sx] == Wx^T (B of GEMM2)
    __shared__ float P_s[8][2][BATCH][16][16];     // per-wave double-buffered TDM tiles (64KB)
    __shared__ float T_s[8][BATCH][16][16];        // per-(wave,channel) intermediate (32KB)

    const int n    = blockIdx.x;
    const int tid  = threadIdx.x;
    const int lane = tid & 31;
    const int wave = tid >> 5;

    // ---- proposal & level selection (uniform per block) ----
    const float x0 = props[n * 7 + 1];
    const float y0 = props[n * 7 + 2];
    const float x1 = props[n * 7 + 3];
    const float y1 = props[n * 7 + 4];

    const float sz = sqrtf((x1 - x0) * (y1 - y0));
    int level = 0;
    {
        const float bases[4] = {8.f, 16.f, 32.f, 64.f};
        float best = fabsf(sz - bases[0]);
        #pragma unroll
        for (int i = 1; i < 4; ++i) {
            float d = fabsf(sz - bases[i]);
            if (d < best) { best = d; level = i; }  // strict '<': first min on ties (argmin)
        }
    }
    const float stride = (float)(2 << level);       // 2,4,8,16
    const int   HW     = 256 >> level;              // 256,128,64,32 (square maps)
    const float* f = (level == 0) ? f0 : (level == 1) ? f1 : (level == 2) ? f2 : f3;

    const float fx0 = x0 / stride, fy0 = y0 / stride;
    const float bw  = (x1 - x0) / stride * (1.0f / CROP);
    const float bh  = (y1 - y0) / stride * (1.0f / CROP);

    // patch origin = clamped floor of first sample (samples are monotonic)
    int xbase, ybase;
    {
        int i0 = (int)floorf(fx0 + 0.5f * bw - 0.5f);
        xbase = min(max(i0, 0), HW - 1);
        int j0 = (int)floorf(fy0 + 0.5f * bh - 0.5f);
        ybase = min(max(j0, 0), HW - 1);
    }

    // ---- build padded interpolation weight matrices in LDS ----
    ((float*)Wy_s)[tid] = 0.0f;
    ((float*)Wx_s)[tid] = 0.0f;
    __syncthreads();

    if (tid < CROP) {                                  // y weights (wave 0)
        int sy = tid;
        float ys = fy0 + (sy + 0.5f) * bh - 0.5f;
        float fl = floorf(ys);
        float w  = ys - fl;
        int i0 = min(max((int)fl, 0), HW - 1);
        int i1 = min(i0 + 1, HW - 1);
        int d0 = min(max(i0 - ybase, 0), 15);
        int d1 = min(max(i1 - ybase, 0), 15);
        Wy_s[sy][d0] += 1.0f - w;
        Wy_s[sy][d1] += w;                             // may alias d0 at border: accumulate
    } else if (tid >= 32 && tid < 32 + CROP) {         // x weights (wave 1)
        int sx = tid - 32;
        float xs = fx0 + (sx + 0.5f) * bw - 0.5f;
        float fl = floorf(xs);
        float w  = xs - fl;
        int i0 = min(max((int)fl, 0), HW - 1);
        int i1 = min(i0 + 1, HW - 1);
        int d0 = min(max(i0 - xbase, 0), 15);
        int d1 = min(max(i1 - xbase, 0), 15);
        Wx_s[d0][sx] += 1.0f - w;
        Wx_s[d1][sx] += w;
    }
    __syncthreads();

    // ---- per-lane constants for WMMA fragment layouts (wave32) ----
    const int m  = lane & 15;     // A: row M / B,D: column N
    const int hi = lane >> 4;     // 0: low half of K (or M), 1: high half
    const uint32_t plane = (uint32_t)HW * (uint32_t)HW;

    const uint32_t rem_x = (uint32_t)(HW - xbase);
    const uint32_t rem_y = (uint32_t)(HW - ybase);
    const size_t   goff  = (size_t)ybase * HW + xbase;
    const int      cbase = wave * 32;

    float* outn = out + (size_t)n * NCH * CROP * CROP;

    // prologue: DMA batch 0 into buffer 0
    tdm_load_patch3d((uint32_t)(uintptr_t)&P_s[wave][0][0][0][0],
                     f + (size_t)cbase * plane + goff,
                     rem_x, rem_y, (uint32_t)(NCH - cbase), (uint32_t)HW, plane);

    for (int b = 0; b < NB; ++b) {
        if (b + 1 < NB) {
            const int cn = cbase + (b + 1) * BATCH;
            tdm_load_patch3d((uint32_t)(uintptr_t)&P_s[wave][(b + 1) & 1][0][0][0],
                             f + (size_t)cn * plane + goff,
                             rem_x, rem_y, (uint32_t)(NCH - cn), (uint32_t)HW, plane);
            // two loads outstanding; in-order completion => oldest (batch b) done at <=1
            asm volatile("s_wait_tensorcnt 0x1" ::: "memory");
        } else {
            asm volatile("s_wait_tensorcnt 0x0" ::: "memory");
        }

        #pragma unroll
        for (int j = 0; j < BATCH; ++j) {
            const int c = cbase + b * BATCH + j;
            float (*P)[16] = P_s[wave][b & 1][j];
            float (*T)[16] = T_s[wave][j];

            // GEMM1: T = Wy * P   (K=16 as 4 chunks of 4)
            v8f acc = {};
            #pragma unroll
            for (int k = 0; k < 4; ++k) {
                int kk = 4 * k + 2 * hi;
                v2f a; a.x = Wy_s[m][kk]; a.y = Wy_s[m][kk + 1];
                v2f bb; bb.x = P[kk][m];  bb.y = P[kk + 1][m];
                acc = __builtin_amdgcn_wmma_f32_16x16x4_f32(
                    false, a, false, bb, (short)0, acc, false, false);
            }

            // spill T (D layout -> LDS row-major) so it can be re-read as A
            #pragma unroll
            for (int r = 0; r < 8; ++r)
                T[r + 8 * hi][m] = acc[r];

            // GEMM2: Out = T * Wx^T
            v8f acc2 = {};
            #pragma unroll
            for (int k = 0; k < 4; ++k) {
                int kk = 4 * k + 2 * hi;
                v2f a; a.x = T[m][kk];     a.y = T[m][kk + 1];
                v2f bb; bb.x = Wx_s[kk][m]; bb.y = Wx_s[kk + 1][m];
                acc2 = __builtin_amdgcn_wmma_f32_16x16x4_f32(
                    false, a, false, bb, (short)0, acc2, false, false);
            }

            // write valid 14x14 region (coalesced along sx)
            float* oc = outn + (size_t)c * CROP * CROP;
            #pragma unroll
            for (int r = 0; r < 8; ++r) {
                int sy = r + 8 * hi;
                if (sy < CROP && m < CROP)
                    oc[sy * CROP + m] = acc2[r];
            }
        }
    }
}

extern "C" void kernel_launch(void* const* d_in, const int* in_sizes, int n_in,
                              void* d_out, int out_size, void* d_ws, size_t ws_size,
                              hipStream_t stream) {
    const float* f0    = (const float*)d_in[0];
    const float* f1    = (const float*)d_in[1];
    const float* f2    = (const float*)d_in[2];
    const float* f3    = (const float*)d_in[3];
    const float* props = (const float*)d_in[4];
    float* out = (float*)d_out;

    const int nroi = in_sizes[4] / 7;   // proposals are [N,7]
    roi_crop_wmma_kernel<<<nroi, 256, 0, stream>>>(f0, f1, f2, f3, props, out);
}